// StockRNN_45887430591111
// MI455X (gfx1250) — compile-verified
//
#include <hip/hip_runtime.h>

// ---------------------------------------------------------------------------
// StockRNN on MI455X (gfx1250): 2-layer ReLU RNN, B=64, T=512, IN=64, H=1024.
//
//   1) pre1 = data @ w_ih0^T + b_ih0        (batched WMMA f32 GEMM, [T,B,H])
//   2) layer-0 recurrence: persistent 16-block kernel, w_hh0 slice pinned in
//      LDS (256KB of 320KB/WGP), grid barrier per timestep, in-place -> ys1
//   3) pre2 = ys1 @ w_ih1^T + b_ih1         (batched WMMA f32 GEMM)
//   4) layer-1 recurrence (same persistent kernel), in-place -> ys2
//   5) fc: out[b*T+t] = dot(ys2[t,b,:], fc_w) + fc_b
//
// All matrix math uses V_WMMA_F32_16X16X4_F32 (native f32 WMMA, full f32
// fidelity over the 512-step serial chain). Inner loops are software-
// pipelined (preload + rotate) so global/LDS loads overlap the WMMA chain,
// and all hot loads/stores use uniform-base + 32-bit offsets (saddr form).
// ---------------------------------------------------------------------------

typedef __attribute__((ext_vector_type(2))) float v2f;
typedef __attribute__((ext_vector_type(8))) float v8f;

#define WMMA_F32X4(a, b, c) \
  __builtin_amdgcn_wmma_f32_16x16x4_f32(false, (a), false, (b), (short)0, (c), false, false)

#define RNN_B      64
#define RNN_T      512
#define RNN_H      1024
#define RNN_NBLK   16           // persistent blocks (one per WGP, 64 cols each)
#define LDS_STRIDE 1028         // padded row stride (floats): 4-bank stagger
#define RNN_LDS_BYTES (64 * LDS_STRIDE * 4)  // 263,168 B < 320KB/WGP

static __device__ __forceinline__ v2f ld2(const float* __restrict__ base, unsigned eoff) {
  return *(const v2f*)(base + eoff);
}

// ---------------------------------------------------------------------------
// init: zero the two grid-barrier counters (one per recurrent layer launch)
// ---------------------------------------------------------------------------
__global__ void rnn_init_kernel(unsigned* ctrs) {
  if (threadIdx.x < 2) ctrs[threadIdx.x] = 0u;
}

// ---------------------------------------------------------------------------
// Batched GEMM: C[m',n] = sum_k A[m,k] * W[n,k] + bias[n]
//   A: [M,K] row-major, W: [N,K] row-major (torch weight layout), N == 1024.
//   perm==1: m = b*512 + t, output row m' = t*64 + b  (data -> [T,B,H])
//   perm==0: output row m' = m                        ([T,B,H] -> [T,B,H])
// One wave computes a 16(M) x 64(N) tile: 4 independent WMMA accumulator
// chains, 4x A-fragment reuse, software-pipelined K loop.
// ---------------------------------------------------------------------------
__global__ void __launch_bounds__(256) gemm_wmma_kernel(
    const float* __restrict__ A, const float* __restrict__ W,
    const float* __restrict__ bias, float* __restrict__ C,
    int M, int N, int K, int perm) {
  const int gw   = blockIdx.x * (blockDim.x >> 5) + (threadIdx.x >> 5);
  const int ngrp = N >> 6;          // N/64 groups
  const int ng   = gw % ngrp;
  const int mt   = gw / ngrp;
  const int lane = threadIdx.x & 31;
  const int lrow = lane & 15;
  const int koff = (lane >> 4) << 1;  // ISA A/B fragment: K={koff,koff+1}

  // 32-bit element offsets from uniform bases -> saddr + voffset addressing.
  const unsigned aoff  = (unsigned)(mt * 16 + lrow) * (unsigned)K + (unsigned)koff;
  const unsigned woff  = (unsigned)(ng * 64 + lrow) * (unsigned)K + (unsigned)koff;
  const unsigned wstep = (unsigned)(16 * K);

  v8f acc0 = v8f{}, acc1 = v8f{}, acc2 = v8f{}, acc3 = v8f{};

  // -- software pipeline: preload kc = 0 --
  v2f a  = ld2(A, aoff);
  v2f b0 = ld2(W, woff);
  v2f b1 = ld2(W, woff + wstep);
  v2f b2 = ld2(W, woff + 2 * wstep);
  v2f b3 = ld2(W, woff + 3 * wstep);

#pragma unroll 4
  for (int kc = 4; kc < K; kc += 4) {
    // issue next-chunk loads before the current WMMA group
    v2f an  = ld2(A, aoff + kc);
    v2f bn0 = ld2(W, woff + kc);
    v2f bn1 = ld2(W, woff + wstep + kc);
    v2f bn2 = ld2(W, woff + 2 * wstep + kc);
    v2f bn3 = ld2(W, woff + 3 * wstep + kc);
    acc0 = WMMA_F32X4(a, b0, acc0);
    acc1 = WMMA_F32X4(a, b1, acc1);
    acc2 = WMMA_F32X4(a, b2, acc2);
    acc3 = WMMA_F32X4(a, b3, acc3);
    a = an; b0 = bn0; b1 = bn1; b2 = bn2; b3 = bn3;
  }
  acc0 = WMMA_F32X4(a, b0, acc0);
  acc1 = WMMA_F32X4(a, b1, acc1);
  acc2 = WMMA_F32X4(a, b2, acc2);
  acc3 = WMMA_F32X4(a, b3, acc3);

  const v8f accs[4] = {acc0, acc1, acc2, acc3};
  const int mhi = (lane >> 4) << 3;   // C/D layout: M = r + 8*(lane>=16)
#pragma unroll
  for (int j = 0; j < 4; ++j) {
    const int ncol = ng * 64 + j * 16 + lrow;
    const float bn = bias[ncol];
#pragma unroll
    for (int r = 0; r < 8; ++r) {
      const int m = mt * 16 + mhi + r;
      unsigned orow;
      if (perm) {
        orow = (unsigned)(m & 511) * 64u + (unsigned)(m >> 9);  // t*B + b
      } else {
        orow = (unsigned)m;
      }
      C[orow * (unsigned)N + (unsigned)ncol] = accs[j][r] + bn;
    }
  }
}

// ---------------------------------------------------------------------------
// Persistent recurrent layer: 16 blocks x 512 threads (16 waves).
// Block bid owns output columns [bid*64, bid*64+64); its 256KB w_hh slice
// lives in LDS for all 512 timesteps. Each wave computes one 16x16 tile of
// Hnew = relu(Pre[t] + Hold @ Whh^T + b_hh). Grid barrier per step via an
// agent-scope release/acquire atomic counter (16 blocks, one per WGP given
// the 263KB LDS footprint -> guaranteed co-resident).
// ---------------------------------------------------------------------------
__global__ void __launch_bounds__(512) rnn_layer_kernel(
    float* pre /* [T,B,H] in: pre-activation, out: ys */,
    const float* __restrict__ whh, const float* __restrict__ bhh,
    const float* __restrict__ h0, float* hbuf /* [2][B*H] ping-pong */,
    float* hlast_out /* [B,H] slot in d_out */, unsigned* ctr) {
  extern __shared__ float lds[];
  const int tid   = threadIdx.x;
  const int nbase = blockIdx.x * 64;
  const unsigned NB = gridDim.x;

  // Stage w_hh[nbase..nbase+63][0..1023] into LDS (padded stride 1028).
  for (int i = tid; i < 64 * 256; i += 512) {
    const int n  = i >> 8;
    const int k4 = i & 255;
    const float4 v = *(const float4*)(whh + (size_t)(nbase + n) * RNN_H + k4 * 4);
    float* d = &lds[n * LDS_STRIDE + k4 * 4];
    d[0] = v.x; d[1] = v.y; d[2] = v.z; d[3] = v.w;
  }
  __syncthreads();

  const int wave = tid >> 5;          // 0..15
  const int lane = tid & 31;
  const int m0   = (wave & 3) * 16;   // 4 m-tiles cover B=64
  const int n0   = (wave >> 2) * 16;  // 4 n-tiles cover the 64-col slice
  const int lrow = lane & 15;
  const int koff = (lane >> 4) << 1;
  const int mhi  = (lane >> 4) << 3;
  const int ncol = nbase + n0 + lrow;

  const float* bptr   = &lds[(n0 + lrow) * LDS_STRIDE + koff];
  const float  bias_n = bhh[ncol];

  // 32-bit element offsets (uniform bases: hold / pre slabs / hbuf halves)
  const unsigned aoff  = (unsigned)(m0 + lrow) * RNN_H + (unsigned)koff;
  const unsigned eoff0 = (unsigned)(m0 + mhi) * RNN_H + (unsigned)ncol;

  for (int t = 0; t < RNN_T; ++t) {
    const float* hold = (t == 0) ? h0 : (hbuf + (size_t)((t - 1) & 1) * RNN_B * RNN_H);

    // -- software pipeline: preload kc = 0 --
    v2f a0 = ld2(hold, aoff);
    v2f b0 = *(const v2f*)(bptr + 0);
    v2f a1 = ld2(hold, aoff + 4);
    v2f b1 = *(const v2f*)(bptr + 4);
    v8f c0 = v8f{}, c1 = v8f{};   // two chains break the WMMA->WMMA RAW

#pragma unroll 4
    for (int kc = 8; kc < RNN_H; kc += 8) {
      v2f a0n = ld2(hold, aoff + kc);
      v2f b0n = *(const v2f*)(bptr + kc);
      v2f a1n = ld2(hold, aoff + kc + 4);
      v2f b1n = *(const v2f*)(bptr + kc + 4);
      c0 = WMMA_F32X4(a0, b0, c0);
      c1 = WMMA_F32X4(a1, b1, c1);
      a0 = a0n; b0 = b0n; a1 = a1n; b1 = b1n;
    }
    c0 = WMMA_F32X4(a0, b0, c0);
    c1 = WMMA_F32X4(a1, b1, c1);
    const v8f c = c0 + c1;

    float* hnew   = hbuf + (size_t)(t & 1) * RNN_B * RNN_H;
    float* prerow = pre + (size_t)t * RNN_B * RNN_H;

    // prefetch next timestep's pre-activation tile (global_prefetch_b8)
    if (t + 1 < RNN_T) {
      __builtin_prefetch(prerow + RNN_B * RNN_H + eoff0, 0, 1);
    }

#pragma unroll
    for (int r = 0; r < 8; ++r) {
      const unsigned idx = eoff0 + (unsigned)r * RNN_H;
      float v = c[r] + prerow[idx] + bias_n;
      v = v > 0.0f ? v : 0.0f;
      prerow[idx] = v;    // ys[t] in place
      hnew[idx]   = v;    // recurrent state
      if (t == RNN_T - 1) hlast_out[idx] = v;
    }

    // ---- grid barrier: release per wave, acquire per thread ----
    __syncthreads();
    if (lane == 0) {
      __hip_atomic_fetch_add(ctr, 1u, __ATOMIC_RELEASE, __HIP_MEMORY_SCOPE_AGENT);
    }
    const unsigned target = (unsigned)(t + 1) * NB * 16u;  // 16 waves/block
    while (__hip_atomic_load(ctr, __ATOMIC_ACQUIRE, __HIP_MEMORY_SCOPE_AGENT) < target) {
      __builtin_amdgcn_s_sleep(2);
    }
    __syncthreads();
  }
}

// ---------------------------------------------------------------------------
// FC: out_data[b*T + t] = dot(ys2[t*B + b, :], fc_w) + fc_b   (OUT == 1)
// One wave per row; float4 loads, shuffle reduction (wave32).
// ---------------------------------------------------------------------------
__global__ void __launch_bounds__(256) fc_kernel(
    const float* __restrict__ ys2, const float* __restrict__ w,
    const float* __restrict__ bsc, float* __restrict__ out) {
  const int row  = blockIdx.x * (blockDim.x >> 5) + (threadIdx.x >> 5);  // 0..32767
  const int lane = threadIdx.x & 31;
  const float4* rp = (const float4*)(ys2 + (size_t)row * RNN_H);
  const float4* wp = (const float4*)w;
  float s = 0.0f;
#pragma unroll
  for (int i = 0; i < 8; ++i) {
    const float4 x = rp[i * 32 + lane];
    const float4 y = wp[i * 32 + lane];
    s += x.x * y.x + x.y * y.y + x.z * y.z + x.w * y.w;
  }
#pragma unroll
  for (int off = 16; off > 0; off >>= 1) s += __shfl_down(s, off, 32);
  if (lane == 0) {
    const int t = row >> 6;   // row = t*B + b
    const int b = row & 63;
    out[b * RNN_T + t] = s + bsc[0];
  }
}

// ---------------------------------------------------------------------------
// Host launcher
// ---------------------------------------------------------------------------
extern "C" void kernel_launch(void* const* d_in, const int* in_sizes, int n_in,
                              void* d_out, int out_size, void* d_ws, size_t ws_size,
                              hipStream_t stream) {
  (void)in_sizes; (void)n_in; (void)out_size; (void)ws_size;
  const float* data   = (const float*)d_in[0];
  const float* hidden = (const float*)d_in[1];
  const float* w_ih0  = (const float*)d_in[2];
  const float* w_hh0  = (const float*)d_in[3];
  const float* b_ih0  = (const float*)d_in[4];
  const float* b_hh0  = (const float*)d_in[5];
  const float* w_ih1  = (const float*)d_in[6];
  const float* w_hh1  = (const float*)d_in[7];
  const float* b_ih1  = (const float*)d_in[8];
  const float* b_hh1  = (const float*)d_in[9];
  const float* fc_w   = (const float*)d_in[10];
  const float* fc_b   = (const float*)d_in[11];
  float* out = (float*)d_out;

  // Workspace layout (floats): [ctrs pad 64] [hbuf 2*B*H] [pre1 T*B*H] [pre2 T*B*H]
  unsigned* ctrs = (unsigned*)d_ws;
  float* wsf  = (float*)d_ws;
  float* hbuf = wsf + 64;
  float* pre1 = hbuf + (size_t)2 * RNN_B * RNN_H;
  float* pre2 = pre1 + (size_t)RNN_T * RNN_B * RNN_H;

  (void)hipFuncSetAttribute((const void*)rnn_layer_kernel,
                            hipFuncAttributeMaxDynamicSharedMemorySize,
                            RNN_LDS_BYTES);

  const int M = RNN_T * RNN_B;                          // 32768
  const int gemm_blocks = (M / 16) * (RNN_H / 64) / 8;  // 4096 (8 waves/block)

  rnn_init_kernel<<<1, 64, 0, stream>>>(ctrs);

  // Layer 0 input projection: [B,T,IN] x [H,IN]^T -> pre1 [T,B,H]
  gemm_wmma_kernel<<<gemm_blocks, 256, 0, stream>>>(
      data, w_ih0, b_ih0, pre1, M, RNN_H, 64, /*perm=*/1);

  // Layer 0 recurrence (in place: pre1 -> ys1), h1 -> d_out hidden slot 0
  rnn_layer_kernel<<<RNN_NBLK, 512, RNN_LDS_BYTES, stream>>>(
      pre1, w_hh0, b_hh0, hidden, hbuf, out + 32768, &ctrs[0]);

  // Layer 1 input projection: ys1 [T,B,H] x [H,H]^T -> pre2 [T,B,H]
  gemm_wmma_kernel<<<gemm_blocks, 256, 0, stream>>>(
      pre1, w_ih1, b_ih1, pre2, M, RNN_H, RNN_H, /*perm=*/0);

  // Layer 1 recurrence (in place: pre2 -> ys2), h2 -> d_out hidden slot 1
  rnn_layer_kernel<<<RNN_NBLK, 512, RNN_LDS_BYTES, stream>>>(
      pre2, w_hh1, b_hh1, hidden + RNN_B * RNN_H, hbuf,
      out + 32768 + RNN_B * RNN_H, &ctrs[1]);

  // Final FC
  fc_kernel<<<gemm_blocks, 256, 0, stream>>>(pre2, fc_w, fc_b, out);
}